// GCNConv_7473243095261
// MI455X (gfx1250) — compile-verified
//
#include <hip/hip_runtime.h>

typedef __attribute__((ext_vector_type(2))) float v2f;
typedef __attribute__((ext_vector_type(8))) float v8f;

#define DOUT 64
#define DIN  64

// ---------------------------------------------------------------------------
// Kernel 0: zero the output accumulator (d_out is poisoned by the harness).
// ---------------------------------------------------------------------------
__global__ void gcn_zero_kernel(float4* __restrict__ p, int n4) {
    int i = blockIdx.x * blockDim.x + threadIdx.x;
    if (i < n4) p[i] = make_float4(0.f, 0.f, 0.f, 0.f);
}

// ---------------------------------------------------------------------------
// Kernel 1: hs = (x @ W) * norm   via V_WMMA_F32_16X16X4_F32.
// One wave32 computes a 16-row x 64-col stripe: 4 accumulators (v8f each),
// K=64 tiled by 4 -> 16 k-steps x 4 col-tiles = 64 WMMAs per wave.
//
// A (16x4 f32) layout: lanes 0-15 hold M=0..15; VGPR0 = K[2*hi], VGPR1 = K[2*hi+1]
//   -> per lane a contiguous k-pair (single 8B load).
// B (4x16 f32) layout mirrors A: v0 = row K=2*hi, v1 = row K=2*hi+1, cols = lane_lo.
// C/D (16x16 f32): VGPR i -> row (i + 8*hi), col = lane_lo.
// ---------------------------------------------------------------------------
__global__ void gcn_gemm_scale_kernel(const float* __restrict__ x,
                                      const float* __restrict__ w,
                                      const float* __restrict__ norm,
                                      float* __restrict__ hs,
                                      int n_tiles) {
    int wave = (int)((blockIdx.x * blockDim.x + threadIdx.x) >> 5);
    if (wave >= n_tiles) return;                  // wave-uniform: EXEC stays all-1s
    int lane    = threadIdx.x & 31;
    int lane_lo = lane & 15;
    int lane_hi = lane >> 4;
    int row0    = wave * 16;

    v8f acc[4] = {v8f{}, v8f{}, v8f{}, v8f{}};

    const float* xrow = x + (size_t)(row0 + lane_lo) * DIN;

    #pragma unroll
    for (int kb = 0; kb < 16; ++kb) {
        int k0 = kb * 4 + lane_hi * 2;
        v2f a;
        a.x = xrow[k0];
        a.y = xrow[k0 + 1];
        #pragma unroll
        for (int cb = 0; cb < 4; ++cb) {
            v2f b;
            b.x = w[(size_t)k0 * DOUT + cb * 16 + lane_lo];
            b.y = w[(size_t)(k0 + 1) * DOUT + cb * 16 + lane_lo];
            acc[cb] = __builtin_amdgcn_wmma_f32_16x16x4_f32(
                /*neg_a=*/false, a, /*neg_b=*/false, b,
                /*c_mod=*/(short)0, acc[cb],
                /*reuse_a=*/false, /*reuse_b=*/false);
        }
    }

    // Epilogue: scale each row by norm[row] (GCN source normalization) and store.
    #pragma unroll
    for (int i = 0; i < 8; ++i) {
        int r = row0 + i + 8 * lane_hi;
        float nv = norm[r];
        #pragma unroll
        for (int cb = 0; cb < 4; ++cb) {
            hs[(size_t)r * DOUT + cb * 16 + lane_lo] = acc[cb][i] * nv;
        }
    }
}

// ---------------------------------------------------------------------------
// Kernel 2: edge scatter. 64 consecutive threads own one edge's 64 channels:
// coalesced gather of hs[src] and global f32 atomic-add into out[dst].
// Accumulation lives in L2 (out = 25.6 MB << 192 MB L2).
// ---------------------------------------------------------------------------
__global__ void gcn_scatter_kernel(const float* __restrict__ hs,
                                   const int* __restrict__ src,
                                   const int* __restrict__ dst,
                                   float* __restrict__ out,
                                   int E) {
    int e = blockIdx.x * 4 + (int)(threadIdx.x >> 6);   // 256 thr = 4 edges
    if (e >= E) return;
    int c = threadIdx.x & 63;
    int s = src[e];
    int d = dst[e];
    float v = hs[(size_t)s * DOUT + c];
    atomicAdd(&out[(size_t)d * DOUT + c], v);
}

// ---------------------------------------------------------------------------
// Kernel 3: finalize in place: out = out * norm[node] + bias[channel].
// ---------------------------------------------------------------------------
__global__ void gcn_finalize_kernel(float* __restrict__ out,
                                    const float* __restrict__ norm,
                                    const float* __restrict__ bias,
                                    int total) {
    int i = blockIdx.x * blockDim.x + threadIdx.x;
    if (i < total) {
        out[i] = out[i] * norm[i >> 6] + bias[i & 63];
    }
}

// ---------------------------------------------------------------------------
// Launch
// ---------------------------------------------------------------------------
extern "C" void kernel_launch(void* const* d_in, const int* in_sizes, int n_in,
                              void* d_out, int out_size, void* d_ws, size_t ws_size,
                              hipStream_t stream) {
    const float* x    = (const float*)d_in[0];   // [N, 64]
    const float* w    = (const float*)d_in[1];   // [64, 64]
    const float* bias = (const float*)d_in[2];   // [64]
    const float* norm = (const float*)d_in[3];   // [N, 1]
    const int*   src  = (const int*)d_in[4];     // [E]
    const int*   dst  = (const int*)d_in[5];     // [E]
    float* out = (float*)d_out;                  // [N, 64]
    float* hs  = (float*)d_ws;                   // [N, 64] scratch (25.6 MB)

    int N = in_sizes[3];                         // norm has N elements
    int E = in_sizes[4];
    int total = N * DOUT;

    // 0) zero accumulator
    int n4 = total / 4;
    gcn_zero_kernel<<<(n4 + 255) / 256, 256, 0, stream>>>((float4*)out, n4);

    // 1) hs = (x @ W) * norm   (WMMA f32)
    int n_tiles = (N + 15) / 16;                 // N=100000 -> 6250 (exact)
    int waves_per_block = 256 / 32;              // 8 waves
    int gemm_blocks = (n_tiles + waves_per_block - 1) / waves_per_block;
    gcn_gemm_scale_kernel<<<gemm_blocks, 256, 0, stream>>>(x, w, norm, hs, n_tiles);

    // 2) scatter-add over edges
    gcn_scatter_kernel<<<(E + 3) / 4, 256, 0, stream>>>(hs, src, dst, out, E);

    // 3) out = out * norm + bias
    gcn_finalize_kernel<<<(total + 255) / 256, 256, 0, stream>>>(out, norm, bias, total);
}